// InvariantPointAttention_8203387535603
// MI455X (gfx1250) — compile-verified
//
#include <hip/hip_runtime.h>
#include <math.h>

#define NRES 768
#define CS_  384
#define CZ_  128
#define CH_  16
#define H_   12
#define PQ_  4
#define PV_  8
#define INF_ 100000.0f
#define EPS_ 1e-8f

typedef __attribute__((ext_vector_type(2))) float v2f;
typedef __attribute__((ext_vector_type(8))) float v8f;
typedef __attribute__((ext_vector_type(4))) int v4i;

__device__ __forceinline__ v8f wmma4(v2f a, v2f b, v8f c) {
  // D = A(16x4,f32) * B(4x16,f32) + C(16x16,f32)
  return __builtin_amdgcn_wmma_f32_16x16x4_f32(false, a, false, b, (short)0, c,
                                               false, false);
}

// ---- CDNA5 async global->LDS path (ASYNCcnt), with safe fallback ----------
#if defined(__has_builtin)
#if __has_builtin(__builtin_amdgcn_global_load_async_to_lds_b128)
#define HAVE_ASYNC_LDS 1
#endif
#endif
#ifndef HAVE_ASYNC_LDS
#define HAVE_ASYNC_LDS 0
#endif

#define GAS1 __attribute__((address_space(1)))
#define LAS3 __attribute__((address_space(3)))

__device__ __forceinline__ void async_copy_b128(const float* __restrict__ g,
                                                float* __restrict__ l) {
#if HAVE_ASYNC_LDS
  __builtin_amdgcn_global_load_async_to_lds_b128((GAS1 v4i*)g, (LAS3 v4i*)l, 0,
                                                 0);
#else
  float4 v = *reinterpret_cast<const float4*>(g);
  *reinterpret_cast<float4*>(l) = v;
#endif
}

#if HAVE_ASYNC_LDS
#if __has_builtin(__builtin_amdgcn_s_wait_asynccnt)
#define WAIT_ASYNC(n) __builtin_amdgcn_s_wait_asynccnt(n)
#else
#define WAIT_ASYNC(n) asm volatile("s_wait_asynccnt %0" ::"i"(n) : "memory")
#endif
#else
#define WAIT_ASYNC(n) \
  do {                \
  } while (0)
#endif

// ---------------------------------------------------------------------------
// K1: projections + rigid transform + WMMA operand packing
//   qcomb[i][h][32] = [ q * 1/sqrt(3*CH) (16) | hw_h * q_pts_glob (12) | 0 (4) ]
//   kcomb[j][h][32] = [ k (16)                | k_pts_glob (12)        | 0 (4) ]
//   vcomb[j][h][48] = [ v (16)                | v_pts_glob (24)        | 0 (8) ]
//   rown[i][h] = -0.5*hw_h*|q_pts|^2 ; coln[j][h] = -0.5*hw_h*|k_pts|^2
// ---------------------------------------------------------------------------
__global__ void __launch_bounds__(256) k_proj(
    const float* __restrict__ s, const float* __restrict__ rot,
    const float* __restrict__ trans,
    const float* __restrict__ wq, const float* __restrict__ bq,
    const float* __restrict__ wkv, const float* __restrict__ bkv,
    const float* __restrict__ wqp, const float* __restrict__ bqp,
    const float* __restrict__ wkvp, const float* __restrict__ bkvp,
    const float* __restrict__ hwts,
    float* __restrict__ qcomb, float* __restrict__ kcomb,
    float* __restrict__ vcomb, float* __restrict__ rown,
    float* __restrict__ coln) {
  __shared__ float srow[CS_];
  __shared__ float hw[H_], qn[H_], kn[H_];
  __shared__ float R[9], T[3];
  const int i = blockIdx.x;
  const int t = threadIdx.x;

  for (int k = t; k < CS_; k += 256) srow[k] = s[(size_t)i * CS_ + k];
  if (t < H_) {
    float x = hwts[t];
    float sp = (x > 20.f) ? x : log1pf(__expf(x));
    hw[t] = sp * 0.13608276348795434f;  // sqrt(1/(3*(PQ*9/2)))
    qn[t] = 0.f;
    kn[t] = 0.f;
  }
  if (t < 9) R[t] = rot[(size_t)i * 9 + t];
  if (t < 3) T[t] = trans[(size_t)i * 3 + t];
  __syncthreads();

  const float sqk = 0.14433756729740643f;  // 1/sqrt(3*CH)

  // q
  for (int o = t; o < H_ * CH_; o += 256) {
    float acc = bq[o];
    const float* w = wq + (size_t)o * CS_;
    for (int k = 0; k < CS_; k++) acc = fmaf(srow[k], w[k], acc);
    int h = o / CH_, c = o % CH_;
    qcomb[((size_t)i * H_ + h) * 32 + c] = acc * sqk;
  }
  // k | v
  for (int o = t; o < 2 * H_ * CH_; o += 256) {
    float acc = bkv[o];
    const float* w = wkv + (size_t)o * CS_;
    for (int k = 0; k < CS_; k++) acc = fmaf(srow[k], w[k], acc);
    int h = o / (2 * CH_), c = o % (2 * CH_);
    if (c < CH_)
      kcomb[((size_t)i * H_ + h) * 32 + c] = acc;
    else
      vcomb[((size_t)i * H_ + h) * 48 + (c - CH_)] = acc;
  }
  // q_pts (rows r = x*H*PQ + h*PQ + p)
  for (int o = t; o < H_ * PQ_; o += 256) {
    int h = o / PQ_, p = o % PQ_;
    float d[3];
#pragma unroll
    for (int x = 0; x < 3; x++) {
      int r = x * (H_ * PQ_) + o;
      float acc = bqp[r];
      const float* w = wqp + (size_t)r * CS_;
      for (int k = 0; k < CS_; k++) acc = fmaf(srow[k], w[k], acc);
      d[x] = acc;
    }
    float g0 = R[0] * d[0] + R[1] * d[1] + R[2] * d[2] + T[0];
    float g1 = R[3] * d[0] + R[4] * d[1] + R[5] * d[2] + T[1];
    float g2 = R[6] * d[0] + R[7] * d[1] + R[8] * d[2] + T[2];
    float sh = hw[h];
    float* dst = qcomb + ((size_t)i * H_ + h) * 32 + 16 + p * 3;
    dst[0] = g0 * sh;
    dst[1] = g1 * sh;
    dst[2] = g2 * sh;
    atomicAdd(&qn[h], g0 * g0 + g1 * g1 + g2 * g2);
  }
  // kv_pts (rows r = x*H*12 + h*12 + p)
  for (int o = t; o < H_ * (PQ_ + PV_); o += 256) {
    int h = o / (PQ_ + PV_), p = o % (PQ_ + PV_);
    float d[3];
#pragma unroll
    for (int x = 0; x < 3; x++) {
      int r = x * (H_ * (PQ_ + PV_)) + o;
      float acc = bkvp[r];
      const float* w = wkvp + (size_t)r * CS_;
      for (int k = 0; k < CS_; k++) acc = fmaf(srow[k], w[k], acc);
      d[x] = acc;
    }
    float g0 = R[0] * d[0] + R[1] * d[1] + R[2] * d[2] + T[0];
    float g1 = R[3] * d[0] + R[4] * d[1] + R[5] * d[2] + T[1];
    float g2 = R[6] * d[0] + R[7] * d[1] + R[8] * d[2] + T[2];
    if (p < PQ_) {
      float* dst = kcomb + ((size_t)i * H_ + h) * 32 + 16 + p * 3;
      dst[0] = g0; dst[1] = g1; dst[2] = g2;
      atomicAdd(&kn[h], g0 * g0 + g1 * g1 + g2 * g2);
    } else {
      float* dst = vcomb + ((size_t)i * H_ + h) * 48 + 16 + (p - PQ_) * 3;
      dst[0] = g0; dst[1] = g1; dst[2] = g2;
    }
  }
  // zero pads
  for (int o = t; o < H_ * 4; o += 256) {
    qcomb[((size_t)i * H_ + o / 4) * 32 + 28 + (o & 3)] = 0.f;
    kcomb[((size_t)i * H_ + o / 4) * 32 + 28 + (o & 3)] = 0.f;
  }
  for (int o = t; o < H_ * 8; o += 256)
    vcomb[((size_t)i * H_ + o / 8) * 48 + 40 + (o & 7)] = 0.f;
  __syncthreads();
  if (t < H_) {
    rown[(size_t)i * H_ + t] = -0.5f * hw[t] * qn[t];
    coln[(size_t)i * H_ + t] = -0.5f * hw[t] * kn[t];
  }
}

// ---------------------------------------------------------------------------
// K2: bias pass (first streaming read of z), WMMA form.
//   Per wave: C(16j x 16h) = z[i, j0:j0+16, :](16x128) @ wb^T(128x16pad)
//   z tile staged through LDS with double-buffered async B128 loads.
//   ab[h][i][j] = (C + bb[h]) / sqrt(3)
// ---------------------------------------------------------------------------
__global__ void __launch_bounds__(256) k_bias(const float* __restrict__ z,
                                              const float* __restrict__ wb,
                                              const float* __restrict__ bb,
                                              float* __restrict__ ab) {
  __shared__ float WBt[CZ_ * 16];          // [k][h(pad16)], 8 KB
  __shared__ float Zb[8][2][16 * 32];      // per-wave double buffer, 32 KB
  const int i = blockIdx.y;
  const int w = threadIdx.x >> 5;
  const int lane = threadIdx.x & 31;
  const int rlo = lane & 15, hi = lane >> 4;
  const int j0 = blockIdx.x * 128 + w * 16;

  for (int t = threadIdx.x; t < CZ_ * 16; t += 256) {
    int k = t >> 4, h = t & 15;
    WBt[t] = (h < H_) ? wb[h * CZ_ + k] : 0.f;
  }
  __syncthreads();

  const float* zrow = z + ((size_t)i * NRES + j0) * CZ_;  // 16 rows x 128

  // stage channels [kc*32, kc*32+32) of the 16 rows: 128 x b128, 4 per lane
#define KB_STAGE(kc, buf)                                           \
  do {                                                              \
    for (int ch = lane; ch < 128; ch += 32) {                       \
      int row_ = ch >> 3, c4_ = ch & 7;                             \
      async_copy_b128(zrow + (size_t)row_ * CZ_ + (kc)*32 + c4_ * 4,\
                      &Zb[w][buf][row_ * 32 + c4_ * 4]);            \
    }                                                               \
  } while (0)

  KB_STAGE(0, 0);
  v8f C = {0.f, 0.f, 0.f, 0.f, 0.f, 0.f, 0.f, 0.f};
  for (int kc = 0; kc < 4; kc++) {
    if (kc < 3) {
      KB_STAGE(kc + 1, (kc + 1) & 1);
      WAIT_ASYNC(4);
    } else {
      WAIT_ASYNC(0);
    }
    const float* Zt = Zb[w][kc & 1];
#pragma unroll
    for (int ks = 0; ks < 8; ks++) {
      int kb = 4 * ks + 2 * hi;
      v2f a, b;
      a.x = Zt[rlo * 32 + kb];          // A[j=rlo][k]
      a.y = Zt[rlo * 32 + kb + 1];
      int kg = kc * 32 + kb;
      b.x = WBt[kg * 16 + rlo];         // B[k][h=rlo]
      b.y = WBt[(kg + 1) * 16 + rlo];
      C = wmma4(a, b, C);
    }
  }
#undef KB_STAGE
  const float sb = 0.5773502691896258f;  // 1/sqrt(3)
  int col = rlo;                         // head
  if (col < H_) {
    float bbv = bb[col];
#pragma unroll
    for (int r = 0; r < 8; r++) {
      int row = r + 8 * hi;              // local j
      ab[((size_t)col * NRES + i) * NRES + (j0 + row)] = sb * (C[r] + bbv);
    }
  }
}

// ---------------------------------------------------------------------------
// K3: attention core. One wave per (head, 16-row tile).
// ---------------------------------------------------------------------------
#define LA_ST 772  // padded stride (768+4) to avoid LDS bank conflicts
#define AT_ST 33   // padded stride for 32-wide fragment tiles
__global__ void __launch_bounds__(32) k_attn(
    const float* __restrict__ qcomb, const float* __restrict__ kcomb,
    const float* __restrict__ vcomb, const float* __restrict__ rown,
    const float* __restrict__ coln, const float* __restrict__ mask,
    float* __restrict__ ab, float* __restrict__ o_out,
    float* __restrict__ optg) {
  __shared__ float La[16 * LA_ST];  // logits -> a
  __shared__ float At[16 * AT_ST];
  __shared__ float Bt[16 * AT_ST];
  __shared__ float Vt[16 * 48];
  __shared__ float rn[16], mi[16];

  const int it = blockIdx.x, h = blockIdx.y;
  const int i0 = it * 16;
  const int lane = threadIdx.x;
  const int rlo = lane & 15, hi = lane >> 4;

  for (int tI = lane; tI < 512; tI += 32)
    At[(tI >> 5) * AT_ST + (tI & 31)] =
        qcomb[((size_t)(i0 + (tI >> 5)) * H_ + h) * 32 + (tI & 31)];
  if (lane < 16) {
    rn[lane] = rown[(size_t)(i0 + lane) * H_ + h];
    mi[lane] = mask[i0 + lane];
  }
  __syncthreads();

  v2f afr[8];
#pragma unroll
  for (int k = 0; k < 8; k++) {
    int kb = 4 * k + 2 * hi;
    afr[k].x = At[rlo * AT_ST + kb];
    afr[k].y = At[rlo * AT_ST + kb + 1];
  }

  // ---- logits, tile by tile over j ----
  for (int jt = 0; jt < NRES / 16; jt++) {
    int j0 = jt * 16;
    for (int tI = lane; tI < 512; tI += 32)
      Bt[(tI >> 5) * AT_ST + (tI & 31)] =
          kcomb[((size_t)(j0 + (tI >> 5)) * H_ + h) * 32 + (tI & 31)];
    __syncthreads();
    v8f C = {0.f, 0.f, 0.f, 0.f, 0.f, 0.f, 0.f, 0.f};
#pragma unroll
    for (int k = 0; k < 8; k++) {
      v2f bfr;
      int kb = 4 * k + 2 * hi;
      bfr.x = Bt[rlo * AT_ST + kb];  // B[kb][col=rlo]
      bfr.y = Bt[rlo * AT_ST + kb + 1];
      C = wmma4(afr[k], bfr, C);
    }
    int col = rlo;
    float cn = coln[(size_t)(j0 + col) * H_ + h];
    float mj = mask[j0 + col];
#pragma unroll
    for (int r = 0; r < 8; r++) {
      int row = r + 8 * hi;
      int gi = i0 + row;
      float bias = ab[((size_t)h * NRES + gi) * NRES + (j0 + col)];
      float lg = C[r] + rn[row] + cn + bias + INF_ * (mi[row] * mj - 1.f);
      La[row * LA_ST + j0 + col] = lg;
    }
    __syncthreads();
  }

  // ---- softmax over j, a -> La and global ab ----
  for (int row = 0; row < 16; row++) {
    float m = -3.4e38f;
    for (int j = lane; j < NRES; j += 32) m = fmaxf(m, La[row * LA_ST + j]);
#pragma unroll
    for (int off = 16; off >= 1; off >>= 1) m = fmaxf(m, __shfl_xor(m, off, 32));
    float ssum = 0.f;
    for (int j = lane; j < NRES; j += 32) {
      float e = __expf(La[row * LA_ST + j] - m);
      La[row * LA_ST + j] = e;
      ssum += e;
    }
#pragma unroll
    for (int off = 16; off >= 1; off >>= 1) ssum += __shfl_xor(ssum, off, 32);
    float inv = 1.f / ssum;
    int gi = i0 + row;
    for (int j = lane; j < NRES; j += 32) {
      float av = La[row * LA_ST + j] * inv;
      La[row * LA_ST + j] = av;
      ab[((size_t)h * NRES + gi) * NRES + j] = av;
    }
  }
  __syncthreads();

  // ---- [o | o_pt_glob] = a @ vcomb : K=768, three 16-wide N tiles ----
  v8f Co  = {0.f, 0.f, 0.f, 0.f, 0.f, 0.f, 0.f, 0.f};
  v8f Cp1 = {0.f, 0.f, 0.f, 0.f, 0.f, 0.f, 0.f, 0.f};
  v8f Cp2 = {0.f, 0.f, 0.f, 0.f, 0.f, 0.f, 0.f, 0.f};
  for (int jt = 0; jt < NRES / 16; jt++) {
    int j0 = jt * 16;
    for (int tI = lane; tI < 768; tI += 32)
      Vt[tI] = vcomb[((size_t)(j0 + tI / 48) * H_ + h) * 48 + (tI % 48)];
    __syncthreads();
#pragma unroll
    for (int ks = 0; ks < 4; ks++) {
      v2f a;
      int kb = j0 + 4 * ks + 2 * hi;
      a.x = La[rlo * LA_ST + kb];
      a.y = La[rlo * LA_ST + kb + 1];
      int kl = 4 * ks + 2 * hi;
      v2f b0, b1, b2;
      b0.x = Vt[kl * 48 + rlo];       b0.y = Vt[(kl + 1) * 48 + rlo];
      b1.x = Vt[kl * 48 + 16 + rlo];  b1.y = Vt[(kl + 1) * 48 + 16 + rlo];
      b2.x = Vt[kl * 48 + 32 + rlo];  b2.y = Vt[(kl + 1) * 48 + 32 + rlo];
      Co  = wmma4(a, b0, Co);
      Cp1 = wmma4(a, b1, Cp1);
      Cp2 = wmma4(a, b2, Cp2);
    }
    __syncthreads();
  }
  int col = rlo;
#pragma unroll
  for (int r = 0; r < 8; r++) {
    int row = r + 8 * hi;
    int gi = i0 + row;
    o_out[((size_t)gi * H_ + h) * CH_ + col] = Co[r];
    optg[((size_t)gi * H_ + h) * 24 + col] = Cp1[r];
    if (col < 8) optg[((size_t)gi * H_ + h) * 24 + 16 + col] = Cp2[r];
  }
}

// ---------------------------------------------------------------------------
// K4: o_pair[i] = a[:,i,:] @ z[i]  (second streaming read of z)
//   a rows staged once per block; z tiles double-buffered via async B128.
//   8 waves per block, wave w handles output columns [16w,16w+16).
// ---------------------------------------------------------------------------
#define AV_ST 772
__global__ void __launch_bounds__(256) k_opair(const float* __restrict__ ab,
                                               const float* __restrict__ z,
                                               float* __restrict__ opair) {
  __shared__ float Av[H_ * AV_ST];   // a rows for this i (~36 KB)
  __shared__ float Zt[2][16 * CZ_];  // 2 x 8 KB z tiles
  const int i = blockIdx.x;
  const int w = threadIdx.x >> 5;
  const int lane = threadIdx.x & 31;
  const int rlo = lane & 15, hi = lane >> 4;

  for (int t = threadIdx.x; t < H_ * NRES; t += 256) {
    int h = t / NRES, j = t % NRES;
    Av[h * AV_ST + j] = ab[((size_t)h * NRES + i) * NRES + j];
  }

  const float* zrow = z + (size_t)i * NRES * CZ_;
#define OP_STAGE(jt, buf)                                            \
  do {                                                               \
    for (int ch = threadIdx.x; ch < 512; ch += 256) {                \
      int row_ = ch >> 5, c4_ = ch & 31;                             \
      async_copy_b128(zrow + (size_t)((jt)*16 + row_) * CZ_ + c4_ * 4,\
                      &Zt[buf][row_ * CZ_ + c4_ * 4]);               \
    }                                                                \
  } while (0)

  OP_STAGE(0, 0);
  __syncthreads();  // Av ready

  const bool avld = (rlo < H_);
  v8f C = {0.f, 0.f, 0.f, 0.f, 0.f, 0.f, 0.f, 0.f};
  for (int jt = 0; jt < NRES / 16; jt++) {
    if (jt < NRES / 16 - 1) {
      OP_STAGE(jt + 1, (jt + 1) & 1);
      WAIT_ASYNC(2);
    } else {
      WAIT_ASYNC(0);
    }
    __syncthreads();
    const float* Zc = Zt[jt & 1];
#pragma unroll
    for (int ks = 0; ks < 4; ks++) {
      int kb = 4 * ks + 2 * hi;
      int kg = jt * 16 + kb;
      v2f a, b;
      a.x = avld ? Av[rlo * AV_ST + kg] : 0.f;
      a.y = avld ? Av[rlo * AV_ST + kg + 1] : 0.f;
      b.x = Zc[kb * CZ_ + w * 16 + rlo];
      b.y = Zc[(kb + 1) * CZ_ + w * 16 + rlo];
      C = wmma4(a, b, C);
    }
    __syncthreads();
  }
#undef OP_STAGE
#pragma unroll
  for (int r = 0; r < 8; r++) {
    int row = r + 8 * hi;
    if (row < H_)
      opair[((size_t)i * H_ + row) * CZ_ + w * 16 + rlo] = C[r];
  }
}

// ---------------------------------------------------------------------------
// K5: inverse-frame rotation of o_pt, norms, concat, output projection.
// ---------------------------------------------------------------------------
__global__ void __launch_bounds__(384) k_final(
    const float* __restrict__ o_in, const float* __restrict__ optg,
    const float* __restrict__ opair, const float* __restrict__ rot,
    const float* __restrict__ trans, const float* __restrict__ wout,
    const float* __restrict__ bout, float* __restrict__ out) {
  __shared__ float cat[2112];
  __shared__ float R[9], T[3];
  const int i = blockIdx.x;
  const int t = threadIdx.x;
  if (t < 9) R[t] = rot[(size_t)i * 9 + t];
  if (t < 3) T[t] = trans[(size_t)i * 3 + t];
  for (int idx = t; idx < H_ * CH_; idx += 384)
    cat[idx] = o_in[(size_t)i * H_ * CH_ + idx];
  for (int idx = t; idx < H_ * CZ_; idx += 384)
    cat[576 + idx] = opair[(size_t)i * H_ * CZ_ + idx];
  __syncthreads();
  for (int idx = t; idx < H_ * PV_; idx += 384) {
    int h = idx / PV_, p = idx % PV_;
    const float* g = optg + ((size_t)i * H_ + h) * 24 + p * 3;
    float g0 = g[0] - T[0], g1 = g[1] - T[1], g2 = g[2] - T[2];
    float lx = R[0] * g0 + R[3] * g1 + R[6] * g2;  // R^T
    float ly = R[1] * g0 + R[4] * g1 + R[7] * g2;
    float lz = R[2] * g0 + R[5] * g1 + R[8] * g2;
    cat[192 + idx] = lx;
    cat[288 + idx] = ly;
    cat[384 + idx] = lz;
    cat[480 + idx] = sqrtf(lx * lx + ly * ly + lz * lz + EPS_);
  }
  __syncthreads();
  for (int col = t; col < CS_; col += 384) {
    const float* w = wout + (size_t)col * 2112;
    float acc = bout[col];
    for (int k = 0; k < 2112; k += 4) {
      float4 wv = *reinterpret_cast<const float4*>(w + k);
      acc = fmaf(cat[k],     wv.x, acc);
      acc = fmaf(cat[k + 1], wv.y, acc);
      acc = fmaf(cat[k + 2], wv.z, acc);
      acc = fmaf(cat[k + 3], wv.w, acc);
    }
    out[(size_t)i * CS_ + col] = acc;
  }
}

extern "C" void kernel_launch(void* const* d_in, const int* in_sizes, int n_in,
                              void* d_out, int out_size, void* d_ws,
                              size_t ws_size, hipStream_t stream) {
  (void)in_sizes; (void)n_in; (void)out_size; (void)ws_size;
  const float* s     = (const float*)d_in[0];
  const float* z     = (const float*)d_in[1];
  const float* rot   = (const float*)d_in[2];
  const float* trans = (const float*)d_in[3];
  const float* mask  = (const float*)d_in[4];
  const float* wq    = (const float*)d_in[5];
  const float* bq    = (const float*)d_in[6];
  const float* wkv   = (const float*)d_in[7];
  const float* bkv   = (const float*)d_in[8];
  const float* wqp   = (const float*)d_in[9];
  const float* bqp   = (const float*)d_in[10];
  const float* wkvp  = (const float*)d_in[11];
  const float* bkvp  = (const float*)d_in[12];
  const float* wb    = (const float*)d_in[13];
  const float* bb    = (const float*)d_in[14];
  const float* hwts  = (const float*)d_in[15];
  const float* wout  = (const float*)d_in[16];
  const float* bout  = (const float*)d_in[17];

  float* ws = (float*)d_ws;
  size_t off = 0;
  float* qcomb = ws + off; off += (size_t)NRES * H_ * 32;
  float* kcomb = ws + off; off += (size_t)NRES * H_ * 32;
  float* vcomb = ws + off; off += (size_t)NRES * H_ * 48;
  float* rown  = ws + off; off += (size_t)NRES * H_;
  float* coln  = ws + off; off += (size_t)NRES * H_;
  float* ab    = ws + off; off += (size_t)H_ * NRES * NRES;  // bias then a (in place)
  float* o_    = ws + off; off += (size_t)NRES * H_ * CH_;
  float* optg  = ws + off; off += (size_t)NRES * H_ * 24;
  float* opair = ws + off; off += (size_t)NRES * H_ * CZ_;

  k_proj<<<NRES, 256, 0, stream>>>(s, rot, trans, wq, bq, wkv, bkv, wqp, bqp,
                                   wkvp, bkvp, hwts, qcomb, kcomb, vcomb, rown,
                                   coln);
  k_bias<<<dim3(NRES / 128, NRES), 256, 0, stream>>>(z, wb, bb, ab);
  k_attn<<<dim3(NRES / 16, H_), 32, 0, stream>>>(qcomb, kcomb, vcomb, rown,
                                                 coln, mask, ab, o_, optg);
  k_opair<<<NRES, 256, 0, stream>>>(ab, z, opair);
  k_final<<<NRES, 384, 0, stream>>>(o_, optg, opair, rot, trans, wout, bout,
                                    (float*)d_out);
}